// Encoder_2104533975457
// MI455X (gfx1250) — compile-verified
//
#include <hip/hip_runtime.h>

typedef __attribute__((ext_vector_type(2))) float v2f;
typedef __attribute__((ext_vector_type(8))) float v8f;

#define K_IN  128
#define K_OUT 64
// LDS row stride (floats) for one k-pair of W: 64 cols * 2 floats + 32 pad.
// 640B stride => hs=0 lanes hit banks 0..31, hs=1 lanes hit banks 32..63 (no conflicts).
#define LWS 160

__device__ __forceinline__ void atomAddF(float* p, float v) {
  unsafeAtomicAdd(p, v);  // global_atomic_add_f32 on gfx1250
}

__device__ __forceinline__ float sigmoidf(float x) {
  return 1.0f / (1.0f + __expf(-x));
}

// ---------------- spectral norm sigma (one power-iteration step) -------------
// v = l2norm(W^T u); s = W v; sigma = ||s||^2 / (||s|| + eps)
__global__ void gcn_sigma_kernel(const float* __restrict__ W,
                                 const float* __restrict__ u,
                                 float* __restrict__ sigma) {
  __shared__ float t[K_OUT];
  __shared__ float red[K_IN];
  const int tid = threadIdx.x;  // 128 threads

  float tv = 0.0f;
  if (tid < K_OUT) {
    #pragma unroll 8
    for (int i = 0; i < K_IN; ++i) tv += W[i * K_OUT + tid] * u[i];
    t[tid] = tv;
  }
  red[tid] = (tid < K_OUT) ? tv * tv : 0.0f;
  __syncthreads();
  for (int s = 64; s > 0; s >>= 1) {
    if (tid < s) red[tid] += red[tid + s];
    __syncthreads();
  }
  const float nt = __fsqrt_rn(red[0]) + 1e-12f;
  __syncthreads();

  float sv = 0.0f;
  #pragma unroll 8
  for (int j = 0; j < K_OUT; ++j) sv += W[tid * K_OUT + j] * t[j];
  sv /= nt;
  red[tid] = sv * sv;
  __syncthreads();
  for (int s = 64; s > 0; s >>= 1) {
    if (tid < s) red[tid] += red[tid + s];
    __syncthreads();
  }
  if (tid == 0) {
    const float ns = __fsqrt_rn(red[0]);
    sigma[0] = (ns * ns) / (ns + 1e-12f);
  }
}

// ---------------- degree: init 1.0 (self loop), scatter sigmoid(ew) ----------
__global__ void gcn_deg_init_kernel(float* __restrict__ deg, int N) {
  const int i = blockIdx.x * 256 + threadIdx.x;
  if (i < N) deg[i] = 1.0f;
}

__global__ void gcn_deg_scatter_kernel(const long long* __restrict__ ei,
                                       const float* __restrict__ ew,
                                       float* __restrict__ deg, int E) {
  const int e = blockIdx.x * 256 + threadIdx.x;
  if (e >= E) return;
  const int col = (int)ei[(size_t)E + e];
  atomAddF(deg + col, sigmoidf(ew[e]));
}

// ---------------- pack per-edge record {row, col, norm_e} --------------------
__global__ void gcn_edge_pack_kernel(const long long* __restrict__ ei,
                                     const float* __restrict__ ew,
                                     const float* __restrict__ deg,
                                     int4* __restrict__ er, int E) {
  const int e = blockIdx.x * 256 + threadIdx.x;
  if (e >= E) return;
  const int row = (int)ei[e];
  const int col = (int)ei[(size_t)E + e];
  const float w = sigmoidf(ew[e]);
  const float nrm = __frsqrt_rn(deg[row]) * w * __frsqrt_rn(deg[col]);
  int4 r;
  r.x = row;
  r.y = col;
  r.z = __float_as_int(nrm);
  r.w = 0;
  er[e] = r;
}

// ---------------- xp = x @ (W / sigma) via V_WMMA_F32_16X16X4_F32 ------------
// 256 threads = 8 waves; each wave owns a 16x64 tile; block covers 128 rows.
// W is staged in LDS pair-interleaved so each B fragment is one ds_load_b64.
__global__ void __launch_bounds__(256)
gcn_gemm_wmma_kernel(const float* __restrict__ x, const float* __restrict__ W,
                     const float* __restrict__ sigmap, float* __restrict__ xp,
                     int Nn) {
  __shared__ float lw2[(K_IN / 2) * LWS];  // 40 KB
  const int tid = threadIdx.x;
  const float inv_sigma = 1.0f / sigmap[0];
  for (int idx = tid; idx < K_IN * K_OUT; idx += 256) {
    const int k = idx >> 6;
    const int col = idx & 63;
    lw2[(k >> 1) * LWS + col * 2 + (k & 1)] = W[idx] * inv_sigma;
  }
  __syncthreads();

  const int wave = tid >> 5;
  const int lane = tid & 31;
  const int lm = lane & 15;  // lane within half
  const int hs = lane >> 4;  // 0 -> K+0/K+1, 1 -> K+2/K+3 (f32 frag layout)
  const int rowBase = blockIdx.x * 128 + wave * 16;

  int m = rowBase + lm;
  if (m >= Nn) m = Nn - 1;  // clamp loads; stores guarded below
  const float* xr = x + (size_t)m * K_IN;

  v8f acc0 = {}, acc1 = {}, acc2 = {}, acc3 = {};
  #pragma unroll 4
  for (int k0 = 0; k0 < K_IN; k0 += 4) {
    const int ka = k0 + hs * 2;                 // even
    const v2f a = *(const v2f*)(xr + ka);       // global_load_b64
    const float* lwp = lw2 + ((k0 >> 1) + hs) * LWS + lm * 2;
    const v2f b0 = *(const v2f*)(lwp +  0);     // ds_load_b64, even pair
    const v2f b1 = *(const v2f*)(lwp + 32);
    const v2f b2 = *(const v2f*)(lwp + 64);
    const v2f b3 = *(const v2f*)(lwp + 96);
    acc0 = __builtin_amdgcn_wmma_f32_16x16x4_f32(false, a, false, b0, (short)0, acc0, false, false);
    acc1 = __builtin_amdgcn_wmma_f32_16x16x4_f32(false, a, false, b1, (short)0, acc1, false, false);
    acc2 = __builtin_amdgcn_wmma_f32_16x16x4_f32(false, a, false, b2, (short)0, acc2, false, false);
    acc3 = __builtin_amdgcn_wmma_f32_16x16x4_f32(false, a, false, b3, (short)0, acc3, false, false);
  }

  // C/D layout: VGPR r -> M = r + 8*hs, N = lm (+16 per tile)
  #pragma unroll
  for (int r = 0; r < 8; ++r) {
    const int row = rowBase + hs * 8 + r;
    if (row < Nn) {
      float* o = xp + (size_t)row * 64 + lm;
      o[0]  = acc0[r];
      o[16] = acc1[r];
      o[32] = acc2[r];
      o[48] = acc3[r];
    }
  }
}

// ---------------- out = xp * (1/deg) + bias (float4 vectorized) --------------
__global__ void gcn_out_init_kernel(const float4* __restrict__ xp4,
                                    const float* __restrict__ deg,
                                    const float4* __restrict__ bias4,
                                    float4* __restrict__ out4, int total4) {
  const int t = blockIdx.x * 256 + threadIdx.x;
  if (t >= total4) return;
  const int i = t >> 4;        // node
  const int j4 = t & 15;       // which float4 of the 64-wide feature
  const float inv = 1.0f / deg[i];
  const float4 v = xp4[t];
  const float4 b = bias4[j4];
  float4 o;
  o.x = v.x * inv + b.x;
  o.y = v.y * inv + b.y;
  o.z = v.z * inv + b.z;
  o.w = v.w * inv + b.w;
  out4[t] = o;
}

// ---------------- scatter-add: out[col] += xp[row] * ne ----------------------
// 16 threads per edge; one b128 record load + one float4 gather + 4 f32 atomics.
__global__ void gcn_scatter_kernel(const int4* __restrict__ er,
                                   const float* __restrict__ xp,
                                   float* __restrict__ out, int E) {
  const int t = blockIdx.x * 256 + threadIdx.x;
  if (t >= E * 16) return;
  const int e = t >> 4;
  const int q = (t & 15) * 4;
  const int4 rec = er[e];
  const float s = __int_as_float(rec.z);
  const float4 v = *(const float4*)(xp + (size_t)rec.x * 64 + q);
  float* dst = out + (size_t)rec.y * 64 + q;
  atomAddF(dst + 0, v.x * s);
  atomAddF(dst + 1, v.y * s);
  atomAddF(dst + 2, v.z * s);
  atomAddF(dst + 3, v.w * s);
}

extern "C" void kernel_launch(void* const* d_in, const int* in_sizes, int n_in,
                              void* d_out, int out_size, void* d_ws, size_t ws_size,
                              hipStream_t stream) {
  const float*     x    = (const float*)d_in[0];
  const long long* ei   = (const long long*)d_in[1];
  const float*     W    = (const float*)d_in[2];
  const float*     bias = (const float*)d_in[3];
  const float*     ew   = (const float*)d_in[4];
  const float*     u    = (const float*)d_in[5];
  float* out = (float*)d_out;

  const int N = in_sizes[0] / K_IN;  // 100000 nodes
  const int E = in_sizes[4];         // 600000 edges

  // workspace layout (floats), all chunks 16B aligned
  float* ws    = (float*)d_ws;
  float* sigma = ws;                               // [1]
  float* deg   = ws + 64;                          // [N]
  size_t erOff = (64 + (size_t)N + 3) & ~(size_t)3;
  int4*  er    = (int4*)(ws + erOff);              // [E] packed {row,col,ne}
  size_t xpOff = erOff + (size_t)E * 4;            // int4 = 4 floats
  float* xp    = ws + xpOff;                       // [N*64]

  gcn_sigma_kernel<<<1, 128, 0, stream>>>(W, u, sigma);
  gcn_deg_init_kernel<<<(N + 255) / 256, 256, 0, stream>>>(deg, N);
  gcn_deg_scatter_kernel<<<(E + 255) / 256, 256, 0, stream>>>(ei, ew, deg, E);
  gcn_edge_pack_kernel<<<(E + 255) / 256, 256, 0, stream>>>(ei, ew, deg, er, E);
  gcn_gemm_wmma_kernel<<<(N + 127) / 128, 256, 0, stream>>>(x, W, sigma, xp, N);
  const int total4 = N * 16;
  gcn_out_init_kernel<<<(total4 + 255) / 256, 256, 0, stream>>>(
      (const float4*)xp, deg, (const float4*)bias, (float4*)out, total4);
  const int scat = E * 16;
  gcn_scatter_kernel<<<(scat + 255) / 256, 256, 0, stream>>>(er, xp, out, E);
}